// GCNModel_77618648973950
// MI455X (gfx1250) — compile-verified
//
#include <hip/hip_runtime.h>

// ---------------------------------------------------------------------------
// 2-layer GCN for MI455X (gfx1250, wave32).
// Dense GEMMs: V_WMMA_F32_16X16X4_F32 (fp32-exact), W staged transposed in LDS,
// one wave per 16-row strip computing TILES_N 16x16 output tiles at once.
// SpMM aggregation: edge-parallel global f32 atomics (all L2-resident).
// ---------------------------------------------------------------------------

typedef __attribute__((ext_vector_type(2))) float v2f;
typedef __attribute__((ext_vector_type(8))) float v8f;

#define GCN_N_NODES 50000
#define GCN_N_EDGES 800000
#define GCN_F_IN    64
#define GCN_F_HID   64
#define GCN_F_OUT   16

#define LDS_KPAD 66   // stride for Wt[n][k]: even -> 8B-aligned ds_load_b64 pairs

// ---------------------------------------------------------------- utilities
__global__ void gcn_zero_f32(float* __restrict__ p, int n) {
    int i = blockIdx.x * blockDim.x + threadIdx.x;
    int stride = gridDim.x * blockDim.x;
    for (; i < n; i += stride) p[i] = 0.0f;
}

__global__ void gcn_degrees(const int* __restrict__ src, const int* __restrict__ dst,
                            float* __restrict__ deg_src, float* __restrict__ deg_dst,
                            int n_edges) {
    int e = blockIdx.x * blockDim.x + threadIdx.x;
    if (e >= n_edges) return;
    atomicAdd(&deg_src[src[e]], 1.0f);
    atomicAdd(&deg_dst[dst[e]], 1.0f);
}

__global__ void gcn_norms(float* __restrict__ deg, int n2) {
    int i = blockIdx.x * blockDim.x + threadIdx.x;
    if (i >= n2) return;
    deg[i] = rsqrtf(fmaxf(deg[i], 1.0f));
}

// --------------------------------------------------------------- WMMA GEMM
// H[m, n] = norm[m] * sum_k X[m,k] * W[k,n]
//
// One wave32 per 16-row strip of H; TILES_N accumulators cover all N columns.
// W (K x N, <= 16KB) is staged TRANSPOSED into LDS once per block so each
// B fragment is a contiguous, 8B-aligned LDS pair -> ds_load_b64.
//
// VGPR layouts assumed (CDNA5 ISA, 32-bit matrices):
//   A 16x4 : lanes 0-15 row M=l hold K={0,1}; lanes 16-31 row M=l hold K={2,3}.
//   B 4x16 : symmetric (lane%16 = column, half selects K pair).
//   C/D    : vgpr j -> row j (lanes 0-15) / row j+8 (lanes 16-31), col lane%16.
template<int TILES_N>
__global__ void gcn_gemm_wmma(const float* __restrict__ X, const float* __restrict__ W,
                              const float* __restrict__ norm_row, float* __restrict__ H,
                              int M, int K, int N) {
    __shared__ float Wt[64][LDS_KPAD];          // Wt[n][k]

    // cooperative transpose-load of W into LDS
    for (int i = threadIdx.x; i < K * N; i += blockDim.x) {
        int kk = i / N;
        int nn = i - kk * N;
        Wt[nn][kk] = W[i];
    }
    __syncthreads();

    int waveInBlk = threadIdx.x >> 5;
    int lane      = threadIdx.x & 31;
    int mtile     = blockIdx.x * (blockDim.x >> 5) + waveInBlk;
    if (mtile >= (M >> 4)) return;              // wave-uniform: EXEC all-ones below

    int m0   = mtile << 4;
    int half = lane >> 4;                       // 0: lanes 0-15, 1: lanes 16-31
    int l    = lane & 15;

    const float* Arow = X + (size_t)(m0 + l) * K;
    v8f acc[TILES_N];
    #pragma unroll
    for (int t = 0; t < TILES_N; ++t) acc[t] = (v8f){};

    for (int k = 0; k < K; k += 4) {
        int ka = k + (half << 1);
        v2f a;
        a.x = Arow[ka + 0];                     // global_load_b64 per lane
        a.y = Arow[ka + 1];
        #pragma unroll
        for (int t = 0; t < TILES_N; ++t) {
            v2f b;
            b.x = Wt[t * 16 + l][ka + 0];       // contiguous pair -> ds_load_b64
            b.y = Wt[t * 16 + l][ka + 1];
            acc[t] = __builtin_amdgcn_wmma_f32_16x16x4_f32(
                         false, a, false, b, (short)0, acc[t], false, false);
        }
    }

    // fused row scaling by norm_src on store
    #pragma unroll
    for (int j = 0; j < 8; ++j) {
        int row = m0 + j + (half << 3);
        float s = norm_row[row];
        #pragma unroll
        for (int t = 0; t < TILES_N; ++t) {
            H[(size_t)row * N + t * 16 + l] = acc[t][j] * s;
        }
    }
}

// --------------------------------------------------------------- SpMM scatter
// AGG[dst[e], :] += H[src[e], :]   — one thread per (edge, 4-feature chunk)
__global__ void gcn_spmm_atomic(const int* __restrict__ src, const int* __restrict__ dst,
                                const float* __restrict__ H, float* __restrict__ AGG,
                                int F, int n_edges) {
    int chunks = F >> 2;
    int tid = blockIdx.x * blockDim.x + threadIdx.x;
    int total = n_edges * chunks;
    if (tid >= total) return;
    int e = tid / chunks;
    int c = (tid - e * chunks) << 2;
    int s = src[e], d = dst[e];
    const float4 v = *(const float4*)(H + (size_t)s * F + c);
    float* p = AGG + (size_t)d * F + c;
    atomicAdd(p + 0, v.x);
    atomicAdd(p + 1, v.y);
    atomicAdd(p + 2, v.z);
    atomicAdd(p + 3, v.w);
}

// --------------------------------------------------------------- epilogue
__global__ void gcn_finalize(const float* __restrict__ AGG, const float* __restrict__ norm_dst,
                             const float* __restrict__ bias, float* __restrict__ OUT,
                             int F, int n_nodes, int do_relu) {
    int i = blockIdx.x * blockDim.x + threadIdx.x;
    if (i >= n_nodes * F) return;
    int row = i / F;
    int col = i - row * F;
    float v = AGG[i] * norm_dst[row] + bias[col];
    if (do_relu) v = fmaxf(v, 0.0f);
    OUT[i] = v;
}

// ---------------------------------------------------------------------------
extern "C" void kernel_launch(void* const* d_in, const int* in_sizes, int n_in,
                              void* d_out, int out_size, void* d_ws, size_t ws_size,
                              hipStream_t stream) {
    const float* features = (const float*)d_in[0];
    const int*   src      = (const int*)  d_in[1];
    const int*   dst      = (const int*)  d_in[2];
    const float* W1       = (const float*)d_in[3];
    const float* b1       = (const float*)d_in[4];
    const float* W2       = (const float*)d_in[5];
    const float* b2       = (const float*)d_in[6];
    float*       out      = (float*)d_out;

    // -------- workspace layout (floats) --------
    float* ws = (float*)d_ws;
    float* H1       = ws;                                    // 50000*64
    float* AGG      = H1  + (size_t)GCN_N_NODES * GCN_F_HID; // 50000*64
    float* norm_src = AGG + (size_t)GCN_N_NODES * GCN_F_HID; // 50000
    float* norm_dst = norm_src + GCN_N_NODES;                // 50000 (contiguous)
    // layer-2 reuse of AGG region:
    float* G2   = AGG;                                       // 50000*16
    float* AGG2 = AGG + (size_t)GCN_N_NODES * GCN_F_OUT;     // 50000*16

    const int TPB = 256;                                     // 8 waves / block

    // 1) degrees -> norms (in place: deg arrays live in norm_src/norm_dst)
    gcn_zero_f32<<<(2 * GCN_N_NODES + TPB - 1) / TPB, TPB, 0, stream>>>(norm_src, 2 * GCN_N_NODES);
    gcn_degrees<<<(GCN_N_EDGES + TPB - 1) / TPB, TPB, 0, stream>>>(src, dst, norm_src, norm_dst, GCN_N_EDGES);
    gcn_norms<<<(2 * GCN_N_NODES + TPB - 1) / TPB, TPB, 0, stream>>>(norm_src, 2 * GCN_N_NODES);

    // 2) layer 1: H1 = (X @ W1) * norm_src   (WMMA f32, 4 N-tiles per wave)
    {
        int strips = GCN_N_NODES / 16;                       // 3125
        int blocks = (strips + 7) / 8;                       // 8 strips per block
        gcn_gemm_wmma<4><<<blocks, TPB, 0, stream>>>(features, W1, norm_src, H1,
                                                     GCN_N_NODES, GCN_F_IN, GCN_F_HID);
    }
    gcn_zero_f32<<<(GCN_N_NODES * GCN_F_HID + TPB - 1) / TPB, TPB, 0, stream>>>(AGG, GCN_N_NODES * GCN_F_HID);
    {
        int total = GCN_N_EDGES * (GCN_F_HID / 4);
        gcn_spmm_atomic<<<(total + TPB - 1) / TPB, TPB, 0, stream>>>(src, dst, H1, AGG, GCN_F_HID, GCN_N_EDGES);
    }
    gcn_finalize<<<(GCN_N_NODES * GCN_F_HID + TPB - 1) / TPB, TPB, 0, stream>>>(
        AGG, norm_dst, b1, H1, GCN_F_HID, GCN_N_NODES, 1);

    // 3) layer 2: G2 = (H1 @ W2) * norm_src   (WMMA f32, 1 N-tile per wave)
    {
        int strips = GCN_N_NODES / 16;
        int blocks = (strips + 7) / 8;
        gcn_gemm_wmma<1><<<blocks, TPB, 0, stream>>>(H1, W2, norm_src, G2,
                                                     GCN_N_NODES, GCN_F_HID, GCN_F_OUT);
    }
    gcn_zero_f32<<<(GCN_N_NODES * GCN_F_OUT + TPB - 1) / TPB, TPB, 0, stream>>>(AGG2, GCN_N_NODES * GCN_F_OUT);
    {
        int total = GCN_N_EDGES * (GCN_F_OUT / 4);
        gcn_spmm_atomic<<<(total + TPB - 1) / TPB, TPB, 0, stream>>>(src, dst, G2, AGG2, GCN_F_OUT, GCN_N_EDGES);
    }
    gcn_finalize<<<(GCN_N_NODES * GCN_F_OUT + TPB - 1) / TPB, TPB, 0, stream>>>(
        AGG2, norm_dst, b2, out, GCN_F_OUT, GCN_N_NODES, 0);
}